// GNNLayer_2714419331120
// MI455X (gfx1250) — compile-verified
//
#include <hip/hip_runtime.h>

typedef __attribute__((ext_vector_type(2))) float v2f;
typedef __attribute__((ext_vector_type(4))) float v4f;
typedef __attribute__((ext_vector_type(8))) float v8f;

#define DF 64

// ---------------------------------------------------------------- zero x ----
__global__ __launch_bounds__(256) void zero_ws_kernel(float* __restrict__ p, int n4) {
  int i = blockIdx.x * 256 + threadIdx.x;
  if (i < n4) {
    v4f z = {0.f, 0.f, 0.f, 0.f};
    *(v4f*)(p + (i << 2)) = z;
  }
}

// ---------------------------------------------- fold the two linears --------
// out2 = x @ (W^T @ W^T) + (b @ W^T + b)  ==  x @ M + c
// M stored pre-swizzled into WMMA-B pair layout: Mp[(i>>1)*128 + j*2 + (i&1)] = M[i][j]
__global__ __launch_bounds__(256) void make_Mc_kernel(const float* __restrict__ W,
                                                      const float* __restrict__ b,
                                                      float* __restrict__ Mp,
                                                      float* __restrict__ c) {
  int t = threadIdx.x;
  for (int e = t; e < DF * DF; e += 256) {
    int i = e >> 6, j = e & 63;
    float s = 0.f;
#pragma unroll 8
    for (int k = 0; k < DF; ++k) s += W[k * DF + i] * W[j * DF + k];  // M[i][j]
    Mp[(i >> 1) * (2 * DF) + j * 2 + (i & 1)] = s;
  }
  if (t < DF) {
    float s = b[t];
#pragma unroll 8
    for (int k = 0; k < DF; ++k) s += b[k] * W[t * DF + k];
    c[t] = s;
  }
}

// ------------------------------------------------------------ COO SpMM ------
// 16 threads per edge, 4 consecutive floats each (b128 gather + 4 f32 atomics)
__global__ __launch_bounds__(256) void spmm_scatter_kernel(
    const float* __restrict__ lap, const int* __restrict__ rows,
    const int* __restrict__ cols, const float* __restrict__ input,
    float* __restrict__ x, int nedges) {
  int gid = blockIdx.x * 256 + threadIdx.x;
  int e = gid >> 4;
  if (e >= nedges) return;
  int q = (gid & 15) << 2;
  int r  = rows[e];
  int ci = cols[e];
  float l = lap[e];
  v4f v = *(const v4f*)(input + (size_t)ci * DF + q);
  float* dst = x + (size_t)r * DF + q;
  unsafeAtomicAdd(dst + 0, l * v.x);   // global_atomic_add_f32, no return
  unsafeAtomicAdd(dst + 1, l * v.y);
  unsafeAtomicAdd(dst + 2, l * v.z);
  unsafeAtomicAdd(dst + 3, l * v.w);
}

// ------------------------------------------------- fused GEMM (WMMA f32) ----
// out[100000x64] = x @ M + c.  One wave32 per 16-row strip, 4 col tiles of 16.
__global__ __launch_bounds__(256) void gemm_fused_kernel(
    const float* __restrict__ x, const float* __restrict__ Mp,
    const float* __restrict__ c, float* __restrict__ out, int nrows) {
  __shared__ float ldsA[128 * DF];  // 8 waves * 16x64 A tiles  (32 KB)
  __shared__ float ldsB[DF * DF];   // M, packed pair layout    (16 KB)
  __shared__ float ldsC[DF];

  int t = threadIdx.x;
  int rowBase0 = blockIdx.x * 128;

  // Block-wide coalesced staging of 128 rows of x into LDS (b128 loads)
#pragma unroll
  for (int j = 0; j < 8; ++j) {
    int fi = (j * 256 + t) << 2;             // float index in 8192-float region
    int row = rowBase0 + (fi >> 6);
    v4f v = {0.f, 0.f, 0.f, 0.f};
    if (row < nrows) v = *(const v4f*)(x + (size_t)rowBase0 * DF + fi);
    *(v4f*)(ldsA + fi) = v;
  }
#pragma unroll
  for (int j = 0; j < 4; ++j) {
    int fi = (j * 256 + t) << 2;
    *(v4f*)(ldsB + fi) = *(const v4f*)(Mp + fi);
  }
  if (t < DF) ldsC[t] = c[t];
  __syncthreads();

  int wave = t >> 5;
  int lane = t & 31;
  int rowBase = rowBase0 + wave * 16;
  if (rowBase >= nrows) return;              // wave-uniform: EXEC stays all-1s
  int m  = lane & 15;                        // A row / B,C column within tile
  int kh = lane >> 4;                        // K half (ISA 16x4 f32 layout)

  const float* A = ldsA + wave * (16 * DF) + m * DF;
  v8f acc0 = {}, acc1 = {}, acc2 = {}, acc3 = {};
#pragma unroll
  for (int k0 = 0; k0 < DF; k0 += 4) {
    int ka = k0 + 2 * kh;
    v2f a = *(const v2f*)(A + ka);                      // A[m][ka], A[m][ka+1]
    const float* Bp = ldsB + (ka >> 1) * (2 * DF);      // pair row base
    v2f b0 = *(const v2f*)(Bp + (m      ) * 2);         // (M[ka][n], M[ka+1][n])
    v2f b1 = *(const v2f*)(Bp + (m + 16 ) * 2);
    v2f b2 = *(const v2f*)(Bp + (m + 32 ) * 2);
    v2f b3 = *(const v2f*)(Bp + (m + 48 ) * 2);
    acc0 = __builtin_amdgcn_wmma_f32_16x16x4_f32(false, a, false, b0, (short)0, acc0, false, false);
    acc1 = __builtin_amdgcn_wmma_f32_16x16x4_f32(false, a, false, b1, (short)0, acc1, false, false);
    acc2 = __builtin_amdgcn_wmma_f32_16x16x4_f32(false, a, false, b2, (short)0, acc2, false, false);
    acc3 = __builtin_amdgcn_wmma_f32_16x16x4_f32(false, a, false, b3, (short)0, acc3, false, false);
  }

  float c0 = ldsC[m], c1 = ldsC[m + 16], c2 = ldsC[m + 32], c3 = ldsC[m + 48];
#pragma unroll
  for (int v = 0; v < 8; ++v) {
    int r = rowBase + v + 8 * kh;            // C/D layout: lanes 16-31 hold M+8
    float* o = out + (size_t)r * DF;
    o[m]      = acc0[v] + c0;
    o[m + 16] = acc1[v] + c1;
    o[m + 32] = acc2[v] + c2;
    o[m + 48] = acc3[v] + c3;
  }
}

// ---------------------------------------------------------------- launch ----
extern "C" void kernel_launch(void* const* d_in, const int* in_sizes, int n_in,
                              void* d_out, int out_size, void* d_ws, size_t ws_size,
                              hipStream_t stream) {
  const float* lap  = (const float*)d_in[0];
  const int*   eidx = (const int*)  d_in[1];
  const float* inp  = (const float*)d_in[2];
  const float* W    = (const float*)d_in[3];
  const float* b    = (const float*)d_in[4];
  float* out = (float*)d_out;

  const int E = in_sizes[0];
  const int N = in_sizes[2] / DF;
  const int* rows = eidx;          // edge_index[0]
  const int* cols = eidx + E;      // edge_index[1]

  // workspace: x [N*64 f32] | Mp [64*64 f32] | c [64 f32]
  float* xbuf = (float*)d_ws;
  float* Mp   = xbuf + (size_t)N * DF;
  float* cvec = Mp + DF * DF;

  int n4 = (N * DF) / 4;
  zero_ws_kernel<<<(n4 + 255) / 256, 256, 0, stream>>>(xbuf, n4);
  make_Mc_kernel<<<1, 256, 0, stream>>>(W, b, Mp, cvec);
  long long tot = (long long)E * 16;
  spmm_scatter_kernel<<<(int)((tot + 255) / 256), 256, 0, stream>>>(lap, rows, cols, inp, xbuf, E);
  int nRowTiles = (N + 15) / 16;
  gemm_fused_kernel<<<(nRowTiles + 7) / 8, 256, 0, stream>>>(xbuf, Mp, cvec, out, N);
}